// PointPillarsPos_71408126263509
// MI455X (gfx1250) — compile-verified
//
#include <hip/hip_runtime.h>
#include <hip/hip_bf16.h>
#include <math.h>
#include <stdint.h>

#define NB       4
#define NN       8192
#define NC       3
#define FEAT     11
#define SCORE_THR 0.1f
#define NMS_THR   0.25f
#define MAXOUT   500
#define THREADS  1024
#define PER_TH   (NN / THREADS)        /* 8  boxes per thread (kernel 1) */
#define CAND     (NC * NN)             /* 24576 candidates per batch      */
#define PER_TH2  (CAND / THREADS)      /* 24 candidates per thread (k2)   */
#define PI_F     3.14159265358979323846f

#if defined(__has_builtin)
#  if __has_builtin(__builtin_amdgcn_global_load_async_to_lds_b32)
#    define USE_ASYNC_LDS 1
#  endif
#endif

#ifdef USE_ASYNC_LDS
typedef __attribute__((address_space(1))) int* gas_int_p;   // global (AS1)
typedef __attribute__((address_space(3))) int* las_int_p;   // LDS    (AS3)
#endif

// ---------------------------------------------------------------------------
// Kernel 1: one workgroup per (batch, class).
//   - stage class scores into LDS (async-to-LDS path on gfx1250)
//   - bitonic sort (score desc, index asc) fully in LDS (49 KB of 320 KB WGP LDS)
//   - greedy NMS: each thread keeps its 8 sorted boxes in VGPRs; keep flags are
//     a 1 KB LDS bitmask; one barrier per live lead, early-exit at first -inf.
//   - emit (score-or--inf, original index) per sorted slot to workspace.
// ---------------------------------------------------------------------------
__global__ __launch_bounds__(THREADS)
void nms_kernel(const float* __restrict__ res,
                float* __restrict__ wsS,
                int*   __restrict__ wsI) {
  __shared__ float          s_key[NN];          // 32 KB
  __shared__ unsigned short s_idx[NN];          // 16 KB
  __shared__ unsigned int   s_keep[NN / 32];    //  1 KB

  const int bc  = blockIdx.x;
  const int b   = bc / NC;
  const int c   = bc % NC;
  const int tid = threadIdx.x;
  const float* base = res + (size_t)b * NN * FEAT;

  // ---- stage raw class scores into LDS --------------------------------
#ifdef USE_ASYNC_LDS
  for (int j = tid; j < NN; j += THREADS) {
    const float* gp = base + (size_t)j * FEAT + 7 + c;
    __builtin_amdgcn_global_load_async_to_lds_b32(
        (gas_int_p)gp, (las_int_p)&s_key[j], 0, 0);
  }
#  if __has_builtin(__builtin_amdgcn_s_wait_asynccnt)
  __builtin_amdgcn_s_wait_asynccnt(0);
#  else
  asm volatile("s_wait_asynccnt 0" ::: "memory");
#  endif
#else
  for (int j = tid; j < NN; j += THREADS) {
    s_key[j] = base[(size_t)j * FEAT + 7 + c];
  }
#endif
  __syncthreads();

  // ---- threshold + init indices ---------------------------------------
  for (int j = tid; j < NN; j += THREADS) {
    float s = s_key[j];
    s_key[j] = (s > SCORE_THR) ? s : -INFINITY;
    s_idx[j] = (unsigned short)j;
  }
  __syncthreads();

  // ---- bitonic sort: score descending, index ascending (stable-like) --
  for (int k = 2; k <= NN; k <<= 1) {
    for (int jj = k >> 1; jj > 0; jj >>= 1) {
      for (int e = tid; e < NN; e += THREADS) {
        int p = e ^ jj;
        if (p > e) {
          bool  up = ((e & k) == 0);
          float ka = s_key[e], kb = s_key[p];
          unsigned short ia = s_idx[e], ib = s_idx[p];
          bool aFirst = (ka > kb) || (ka == kb && ia < ib);
          if (up != aFirst) {
            s_key[e] = kb; s_key[p] = ka;
            s_idx[e] = ib; s_idx[p] = ia;
          }
        }
      }
      __syncthreads();
    }
  }

  // ---- init keep bitmask (valid = finite score) -----------------------
  for (int w = tid; w < NN / 32; w += THREADS) {
    unsigned int m = 0u;
    for (int t = 0; t < 32; ++t)
      if (s_key[w * 32 + t] != -INFINITY) m |= (1u << t);
    s_keep[w] = m;
  }

  // ---- load my 8 sorted boxes into VGPRs ------------------------------
  float x1[PER_TH], y1[PER_TH], x2[PER_TH], y2[PER_TH], ar[PER_TH];
  int   myKeep = 0;
  const int jbase = tid * PER_TH;
#pragma unroll
  for (int r = 0; r < PER_TH; ++r) {
    int orig = s_idx[jbase + r];
    const float* bp = base + (size_t)orig * FEAT;
    float x = bp[0], y = bp[1], l = bp[3], w = bp[4];
    x1[r] = x - 0.5f * l;  x2[r] = x + 0.5f * l;
    y1[r] = y - 0.5f * w;  y2[r] = y + 0.5f * w;
    ar[r] = l * w;
    if (s_key[jbase + r] != -INFINITY) myKeep |= (1 << r);
  }
  __syncthreads();

  // ---- greedy NMS over sorted leads -----------------------------------
  for (int i = 0; i < NN; ++i) {
    float si = s_key[i];                 // uniform -> uniform break
    if (si == -INFINITY) break;          // rest are invalid (sorted)
    bool alive = (s_keep[i >> 5] >> (i & 31)) & 1u;
    if (alive && myKeep) {
      int orig = s_idx[i];
      const float* bp = base + (size_t)orig * FEAT;
      if (i + 1 < NN) {                  // prefetch next lead's row
        __builtin_prefetch(base + (size_t)s_idx[i + 1] * FEAT, 0, 0);
      }
      float xi = bp[0], yi = bp[1], li = bp[3], wi = bp[4];
      float ix1 = xi - 0.5f * li, ix2 = xi + 0.5f * li;
      float iy1 = yi - 0.5f * wi, iy2 = yi + 0.5f * wi;
      float ia  = li * wi;
#pragma unroll
      for (int r = 0; r < PER_TH; ++r) {
        int j = jbase + r;
        if (j > i && ((myKeep >> r) & 1)) {
          float iw = fminf(ix2, x2[r]) - fmaxf(ix1, x1[r]);
          float ih = fminf(iy2, y2[r]) - fmaxf(iy1, y1[r]);
          iw = fmaxf(iw, 0.0f);
          ih = fmaxf(ih, 0.0f);
          float inter = iw * ih;
          float iou   = inter / (ia + ar[r] - inter + 1e-8f);
          if (iou > NMS_THR) {
            myKeep &= ~(1 << r);
            atomicAnd(&s_keep[j >> 5], ~(1u << (j & 31)));
          }
        }
      }
    }
    __syncthreads();
  }

  // ---- emit per-class candidates --------------------------------------
#pragma unroll
  for (int r = 0; r < PER_TH; ++r) {
    int j = jbase + r;
    float s = ((myKeep >> r) & 1) ? s_key[j] : -INFINITY;
    size_t o = (size_t)b * CAND + (size_t)c * NN + j;
    wsS[o] = s;
    wsI[o] = (int)s_idx[j];
  }
}

// ---------------------------------------------------------------------------
// Kernel 2: one workgroup per batch. 24576 candidates, exactly 24 per thread
// held in VGPRs. 500 iterations of a wave32 shfl_xor argmax tree
// (score desc, candidate-index asc tie-break == jax.lax.top_k order),
// then threads 0..499 gather and write boxes / labels / scores.
// ---------------------------------------------------------------------------
__global__ __launch_bounds__(THREADS)
void topk_kernel(const float* __restrict__ res,
                 const float* __restrict__ wsS,
                 const int*   __restrict__ wsI,
                 float* __restrict__ outBoxes,
                 int*   __restrict__ outLabels,
                 float* __restrict__ outScores) {
  __shared__ float redS[32];
  __shared__ int   redQ[32];
  __shared__ float selS[MAXOUT];
  __shared__ int   selQ[MAXOUT];

  const int b    = blockIdx.x;
  const int tid  = threadIdx.x;
  const int lane = tid & 31;
  const int wv   = tid >> 5;
  const float* S = wsS + (size_t)b * CAND;

  float sc[PER_TH2];
#pragma unroll
  for (int s = 0; s < PER_TH2; ++s) sc[s] = S[tid + s * THREADS];
  unsigned int used = 0u;

  for (int k = 0; k < MAXOUT; ++k) {
    // per-thread argmax over unconsumed slots (slot s -> q = tid + s*1024)
    float bs = -INFINITY;
    int   bq = 0x7fffffff;
#pragma unroll
    for (int s = 0; s < PER_TH2; ++s) {
      if (!((used >> s) & 1u)) {
        float v = sc[s];
        int   q = tid + s * THREADS;
        if (v > bs || (v == bs && q < bq)) { bs = v; bq = q; }
      }
    }
    // wave32 reduction
#pragma unroll
    for (int off = 16; off > 0; off >>= 1) {
      float os = __shfl_xor(bs, off, 32);
      int   oq = __shfl_xor(bq, off, 32);
      if (os > bs || (os == bs && oq < bq)) { bs = os; bq = oq; }
    }
    if (lane == 0) { redS[wv] = bs; redQ[wv] = bq; }
    __syncthreads();
    if (wv == 0) {
      bs = redS[lane]; bq = redQ[lane];
#pragma unroll
      for (int off = 16; off > 0; off >>= 1) {
        float os = __shfl_xor(bs, off, 32);
        int   oq = __shfl_xor(bq, off, 32);
        if (os > bs || (os == bs && oq < bq)) { bs = os; bq = oq; }
      }
      if (lane == 0) { selS[k] = bs; selQ[k] = bq; }
    }
    __syncthreads();
    int wq = selQ[k];
    if ((wq & (THREADS - 1)) == tid) used |= 1u << (wq >> 10);
  }
  __syncthreads();

  // gather + write final outputs
  for (int k = tid; k < MAXOUT; k += THREADS) {
    int   q = selQ[k];
    float s = selS[k];
    int   c = q / NN;
    int orig = wsI[(size_t)b * CAND + q];
    const float* bp = res + ((size_t)b * NN + orig) * FEAT;
    float x = bp[0], y = bp[1], z = bp[2];
    float l = bp[3], w = bp[4], h = bp[5];
    float th = bp[6];
    float dir = bp[10];
    // limit_period(th) + (1-dir)*pi
    float ang = th - floorf(th / PI_F + 1.0f) * PI_F + (1.0f - dir) * PI_F;
    float* ob = outBoxes + ((size_t)b * MAXOUT + k) * 7;
    ob[0] = x; ob[1] = y; ob[2] = z;
    ob[3] = l; ob[4] = w; ob[5] = h;
    ob[6] = ang;
    outLabels[b * MAXOUT + k] = c;
    outScores[b * MAXOUT + k] = s;
  }
}

// ---------------------------------------------------------------------------
extern "C" void kernel_launch(void* const* d_in, const int* in_sizes, int n_in,
                              void* d_out, int out_size, void* d_ws, size_t ws_size,
                              hipStream_t stream) {
  (void)in_sizes; (void)n_in; (void)out_size; (void)ws_size;
  const float* res = (const float*)d_in[0];        // (4, 8192, 11) f32
  float* out = (float*)d_out;                      // 18000 elements

  // workspace: per-batch candidate scores then indices (768 KB total)
  float* wsS = (float*)d_ws;
  int*   wsI = (int*)((char*)d_ws + (size_t)NB * CAND * sizeof(float));

  float* outBoxes  = out;                                   // 4*500*7
  int*   outLabels = (int*)(out + (size_t)NB * MAXOUT * 7); // 4*500 (i32 bits)
  float* outScores = out + (size_t)NB * MAXOUT * 7 + (size_t)NB * MAXOUT;

  nms_kernel<<<NB * NC, THREADS, 0, stream>>>(res, wsS, wsI);
  topk_kernel<<<NB, THREADS, 0, stream>>>(res, wsS, wsI,
                                          outBoxes, outLabels, outScores);
}